// LightGCN_75900662055226
// MI455X (gfx1250) — compile-verified
//
#include <hip/hip_runtime.h>
#include <hip/hip_bf16.h>

typedef float v2f __attribute__((ext_vector_type(2)));
typedef float v8f __attribute__((ext_vector_type(8)));

#if defined(__HIP_DEVICE_COMPILE__) && __has_builtin(__builtin_amdgcn_wmma_f32_16x16x4_f32)
#define WMMA_AVAIL 1
#else
#define WMMA_AVAIL 0
#endif

#define WPB 8  // waves per block in the layer kernel

// ---------------- degree count (int atomics: deterministic) ----------------
__global__ void k_count_deg(const int* __restrict__ dst, int* __restrict__ deg, int E) {
  int e = blockIdx.x * blockDim.x + threadIdx.x;
  if (e < E) atomicAdd(&deg[dst[e]], 1);
}

__global__ void k_dinv(const int* __restrict__ deg, float* __restrict__ dinv, int N) {
  int i = blockIdx.x * blockDim.x + threadIdx.x;
  if (i < N) {
    int d = deg[i];
    dinv[i] = (d > 0) ? (1.0f / sqrtf((float)d)) : 0.0f;
  }
}

// ---------------- 3-phase exclusive scan of deg -> row_ptr ----------------
__global__ void k_scan_block(const int* __restrict__ deg, int* __restrict__ rp,
                             int* __restrict__ bsum, int N) {
  __shared__ int s[256];
  int tid = threadIdx.x;
  int i = blockIdx.x * 256 + tid;
  int v = (i < N) ? deg[i] : 0;
  s[tid] = v;
  __syncthreads();
  for (int off = 1; off < 256; off <<= 1) {
    int t = (tid >= off) ? s[tid - off] : 0;
    __syncthreads();
    s[tid] += t;
    __syncthreads();
  }
  if (i < N) rp[i] = s[tid] - v;  // exclusive within block
  if (tid == 255) bsum[blockIdx.x] = s[255];
}

__global__ void k_scan_spine(int* __restrict__ bsum, int* __restrict__ rp, int nb, int N) {
  if (blockIdx.x == 0 && threadIdx.x == 0) {
    int run = 0;
    for (int b = 0; b < nb; ++b) { int t = bsum[b]; bsum[b] = run; run += t; }
    rp[N] = run;  // total = E
  }
}

__global__ void k_scan_add(int* __restrict__ rp, const int* __restrict__ bsum, int N) {
  int i = blockIdx.x * blockDim.x + threadIdx.x;
  if (i < N) rp[i] += bsum[i >> 8];
}

// ---------------- CSR fill (dst-grouped) ----------------
__global__ void k_csr_fill(const int* __restrict__ src, const int* __restrict__ dst,
                           const float* __restrict__ dinv, int* __restrict__ cursor,
                           int* __restrict__ col, float* __restrict__ val, int E) {
  int e = blockIdx.x * blockDim.x + threadIdx.x;
  if (e < E) {
    int s = src[e], d = dst[e];
    int pos = atomicAdd(&cursor[d], 1);
    col[pos] = s;
    val[pos] = dinv[s] * dinv[d];
  }
}

// ---------------- one propagation layer: x_new = A_hat * x, acc += x_new ----
// One wave owns one 16-row destination tile. Edge groups of 4 map onto
// V_WMMA_F32_16X16X4_F32: A(16x4)=one-hot weights, B(4x16)=gathered src rows,
// D(16x16 f32)=tile accumulator. 4 WMMAs per group cover all 64 features.
__global__ __launch_bounds__(WPB * 32) void k_layer(
    const float* __restrict__ x, const int* __restrict__ rp,
    const int* __restrict__ col, const float* __restrict__ val,
    float* __restrict__ xnew, float* __restrict__ acc, int N, int ntiles) {
  __shared__ int sb[WPB][17];
#if !WMMA_AVAIL
  __shared__ float fax[WPB][32], fay[WPB][32];
  __shared__ float fbx[WPB][4][32], fby[WPB][4][32];
#endif
  const int wslot = threadIdx.x >> 5;
  const int lane  = threadIdx.x & 31;
  const int tile  = blockIdx.x * WPB + wslot;
  if (tile >= ntiles) return;  // wave-uniform: waves that proceed have EXEC all-1s

  if (lane < 17) {
    int idx = tile * 16 + lane;
    sb[wslot][lane] = rp[(idx <= N) ? idx : N];
  }
  const int base = sb[wslot][0];
  const int end  = sb[wslot][16];
  const int half = lane >> 4;   // K-pair select (edges 0,1 vs 2,3)
  const int n    = lane & 15;   // A row index m / B col index n for this lane

  v8f c0 = {}, c1 = {}, c2 = {}, c3 = {};

  for (int p = base; p < end; p += 4) {
    __builtin_prefetch(col + p + 4, 0, 1);
    int eA = p + half * 2;
    int eB = eA + 1;
    int eAc = (eA < end) ? eA : (end - 1);   // clamp: keep loads unconditional,
    int eBc = (eB < end) ? eB : (end - 1);   // mask via zero weight (EXEC stays full)
    int   sA = col[eAc];
    int   sB = col[eBc];
    float wAv = val[eAc];
    float wBv = val[eBc];
    float wA = (eA < end) ? wAv : 0.0f;
    float wB = (eB < end) ? wBv : 0.0f;
    // local dst row of each edge = #tile-row-boundaries <= edge position
    int dA = 0, dB = 0;
#pragma unroll
    for (int r = 1; r < 16; ++r) {
      dA += (eAc >= sb[wslot][r]) ? 1 : 0;
      dB += (eBc >= sb[wslot][r]) ? 1 : 0;
    }
    // A(16x4) layout: lanes 0-15 hold K=0,1 of row m=lane; lanes 16-31 hold K=2,3
    v2f a;
    a.x = (dA == n) ? wA : 0.0f;
    a.y = (dB == n) ? wB : 0.0f;
    // gather the two source rows this half-wave is responsible for
    const float* rowA = x + (size_t)sA * 64;
    const float* rowB = x + (size_t)sB * 64;
    float bA[4], bB[4];
#pragma unroll
    for (int c = 0; c < 4; ++c) { bA[c] = rowA[c * 16 + n]; bB[c] = rowB[c * 16 + n]; }

#if WMMA_AVAIL
    {
      v2f b = {bA[0], bB[0]};
      c0 = __builtin_amdgcn_wmma_f32_16x16x4_f32(false, a, false, b, (short)0, c0, false, false);
    }
    {
      v2f b = {bA[1], bB[1]};
      c1 = __builtin_amdgcn_wmma_f32_16x16x4_f32(false, a, false, b, (short)0, c1, false, false);
    }
    {
      v2f b = {bA[2], bB[2]};
      c2 = __builtin_amdgcn_wmma_f32_16x16x4_f32(false, a, false, b, (short)0, c2, false, false);
    }
    {
      v2f b = {bA[3], bB[3]};
      c3 = __builtin_amdgcn_wmma_f32_16x16x4_f32(false, a, false, b, (short)0, c3, false, false);
    }
#else
    // LDS mini-GEMM fallback producing the identical D layout
    fax[wslot][lane] = a.x;
    fay[wslot][lane] = a.y;
#pragma unroll
    for (int c = 0; c < 4; ++c) { fbx[wslot][c][lane] = bA[c]; fby[wslot][c][lane] = bB[c]; }
#pragma unroll
    for (int r = 0; r < 8; ++r) {
      int m = half * 8 + r;
      float a0 = fax[wslot][m],      a1 = fay[wslot][m];
      float a2 = fax[wslot][m + 16], a3 = fay[wslot][m + 16];
      c0[r] += a0 * fbx[wslot][0][n] + a1 * fby[wslot][0][n] + a2 * fbx[wslot][0][n + 16] + a3 * fby[wslot][0][n + 16];
      c1[r] += a0 * fbx[wslot][1][n] + a1 * fby[wslot][1][n] + a2 * fbx[wslot][1][n + 16] + a3 * fby[wslot][1][n + 16];
      c2[r] += a0 * fbx[wslot][2][n] + a1 * fby[wslot][2][n] + a2 * fbx[wslot][2][n + 16] + a3 * fby[wslot][2][n + 16];
      c3[r] += a0 * fbx[wslot][3][n] + a1 * fby[wslot][3][n] + a2 * fbx[wslot][3][n + 16] + a3 * fby[wslot][3][n + 16];
    }
#endif
  }

  // D layout: VGPR r -> M=r (lanes 0-15), M=r+8 (lanes 16-31); N = lane&15 per chunk
#pragma unroll
  for (int r = 0; r < 8; ++r) {
    int row = tile * 16 + half * 8 + r;
    if (row < N) {
      size_t idx = (size_t)row * 64 + n;
      xnew[idx]      = c0[r];  acc[idx]      += c0[r];
      xnew[idx + 16] = c1[r];  acc[idx + 16] += c1[r];
      xnew[idx + 32] = c2[r];  acc[idx + 32] += c2[r];
      xnew[idx + 48] = c3[r];  acc[idx + 48] += c3[r];
    }
  }
}

__global__ void k_scale(float* __restrict__ out, float s, int n) {
  int i = blockIdx.x * blockDim.x + threadIdx.x;
  if (i < n) out[i] *= s;
}

extern "C" void kernel_launch(void* const* d_in, const int* in_sizes, int n_in,
                              void* d_out, int out_size, void* d_ws, size_t ws_size,
                              hipStream_t stream) {
  (void)n_in; (void)out_size; (void)ws_size;
  const int*   edge_index = (const int*)d_in[0];
  const float* emb        = (const float*)d_in[1];
  const int E = in_sizes[0] / 2;
  const int N = in_sizes[1] / 64;
  const int num_layers = 3;
  const int* src = edge_index;
  const int* dst = edge_index + E;

  char* ws = (char*)d_ws;
  size_t off = 0;
  auto alloc = [&](size_t bytes) -> void* {
    void* p = ws + off;
    off = (off + bytes + 255) & ~(size_t)255;
    return p;
  };
  int*   deg    = (int*)  alloc((size_t)N * 4);
  float* dinv   = (float*)alloc((size_t)N * 4);
  int*   rp     = (int*)  alloc((size_t)(N + 1) * 4);
  int*   cursor = (int*)  alloc((size_t)N * 4);
  int*   bsum   = (int*)  alloc((size_t)4096 * 4);
  int*   col    = (int*)  alloc((size_t)E * 4);
  float* val    = (float*)alloc((size_t)E * 4);
  float* xA     = (float*)alloc((size_t)N * 64 * 4);
  float* xB     = (float*)alloc((size_t)N * 64 * 4);
  float* out    = (float*)d_out;

  // 1) degrees + normalization
  hipMemsetAsync(deg, 0, (size_t)N * 4, stream);
  k_count_deg<<<(E + 255) / 256, 256, 0, stream>>>(dst, deg, E);
  k_dinv<<<(N + 255) / 256, 256, 0, stream>>>(deg, dinv, N);

  // 2) exclusive scan -> row_ptr
  int nb = (N + 255) / 256;
  k_scan_block<<<nb, 256, 0, stream>>>(deg, rp, bsum, N);
  k_scan_spine<<<1, 32, 0, stream>>>(bsum, rp, nb, N);
  k_scan_add<<<nb, 256, 0, stream>>>(rp, bsum, N);

  // 3) CSR (edges grouped by destination)
  hipMemcpyAsync(cursor, rp, (size_t)N * 4, hipMemcpyDeviceToDevice, stream);
  k_csr_fill<<<(E + 255) / 256, 256, 0, stream>>>(src, dst, dinv, cursor, col, val, E);

  // 4) acc = emb ; x0 = emb
  hipMemcpyAsync(xA, emb, (size_t)N * 64 * 4, hipMemcpyDeviceToDevice, stream);
  hipMemcpyAsync(out, emb, (size_t)N * 64 * 4, hipMemcpyDeviceToDevice, stream);

  // 5) three WMMA propagation layers (atomic-free, L2-resident gathers)
  int ntiles  = (N + 15) / 16;
  int lblocks = (ntiles + WPB - 1) / WPB;
  float* xcur = xA;
  float* xnext = xB;
  for (int l = 0; l < num_layers; ++l) {
    k_layer<<<lblocks, WPB * 32, 0, stream>>>(xcur, rp, col, val, xnext, out, N, ntiles);
    float* t = xcur; xcur = xnext; xnext = t;
  }

  // 6) final average over (num_layers + 1)
  int total = N * 64;
  k_scale<<<(total + 255) / 256, 256, 0, stream>>>(out, 1.0f / (float)(num_layers + 1), total);
}